// ZBL_potential_38714835206648
// MI455X (gfx1250) — compile-verified
//
#include <hip/hip_runtime.h>
#include <hip/hip_bf16.h>

// ---------------------------------------------------------------------------
// ZBL potential: edge-parallel gather/compute/scatter + segmented reductions.
// Memory-bound irregular kernel -> no WMMA; CDNA5 paths used instead:
//   * global_load_async_to_lds_b128 + s_wait_asynccnt  (per-wave double buffer)
//   * ds_add_f32 LDS atomics for the graph reduction
//   * native global_atomic_add_f32 (unsafeAtomicAdd) for node scatter
// ---------------------------------------------------------------------------

__global__ void zbl_zero_kernel(float* __restrict__ p, int n) {
    int i = blockIdx.x * blockDim.x + threadIdx.x;
    if (i < n) p[i] = 0.0f;
}

__device__ __forceinline__ void zbl_edge_accum(
    const float* __restrict__ coords, const float* __restrict__ species,
    float* __restrict__ node_e, int s, int d, float shx, float shy, float shz)
{
    float zi = species[s];
    float zj = species[d];
    const float* cs = coords + 3 * (size_t)s;
    const float* cd = coords + 3 * (size_t)d;
    float dx = cs[0] - (cd[0] + shx);
    float dy = cs[1] - (cd[1] + shy);
    float dz = cs[2] - (cd[2] + shz);
    float dist = sqrtf(dx * dx + dy * dy + dz * dz) + 1e-9f;
    // a = 0.4543 / (Zi^0.3 + Zj^0.3);  x = r / a
    float denom = __powf(zi, 0.3f) + __powf(zj, 0.3f);
    float a = 0.4543f / denom;
    float x = dist / a;
    float phi = 0.1818f   * __expf(-3.2f    * x)
              + 0.5099f   * __expf(-0.9423f * x)
              + 0.2802f   * __expf(-0.4028f * x)
              + 0.02817f  * __expf(-0.2016f * x);
    float ee = (14.3996f * zi * zj / dist) * phi;
    unsafeAtomicAdd(&node_e[s], ee);   // global_atomic_add_f32, no return
}

// 256 threads = 8 waves. Each wave processes tiles of 128 edges: lane l owns
// edges tile*128 + 4l .. +3. src/dst index words are staged into a private
// per-wave LDS double buffer via async DMA; each wave only reads what its own
// async ops wrote, so the only sync needed is s_wait_asynccnt (no barriers).
__global__ __launch_bounds__(256) void zbl_edge_kernel(
    const float* __restrict__ coords, const float* __restrict__ species,
    const int* __restrict__ ei, const float* __restrict__ shifts,
    float* __restrict__ node_e, int E)
{
    __shared__ __align__(16) char s_stage[8 * 2048]; // 8 waves * 2 bufs * (512B src + 512B dst)
    const unsigned lds_base = (unsigned)(unsigned long long)(const void*)&s_stage[0];
    const int lane  = threadIdx.x & 31;
    const int wave  = threadIdx.x >> 5;
    const int gwave = blockIdx.x * 8 + wave;
    const int nwaves = gridDim.x * 8;
    const int full_tiles = E >> 7;   // 128 edges per tile

    auto prefetch = [&](int tile, int buf) {
        unsigned l_src = lds_base + (unsigned)(wave * 2048 + buf * 1024 + lane * 16);
        unsigned l_dst = l_src + 512u;
        unsigned long long g_src = (unsigned long long)(const void*)ei
                                 + ((unsigned long long)tile * 128ull + (unsigned)(lane * 4)) * 4ull;
        unsigned long long g_dst = g_src + (unsigned long long)E * 4ull;
        asm volatile("global_load_async_to_lds_b128 %0, %1, off"
                     :: "v"(l_src), "v"(g_src) : "memory");
        asm volatile("global_load_async_to_lds_b128 %0, %1, off"
                     :: "v"(l_dst), "v"(g_dst) : "memory");
    };

    int t = gwave;
    int buf = 0;
    if (t < full_tiles) prefetch(t, buf);
    for (; t < full_tiles; t += nwaves) {
        int nt = t + nwaves;
        if (nt < full_tiles) {
            prefetch(nt, buf ^ 1);
            asm volatile("s_wait_asynccnt 2" ::: "memory"); // current tile's 2 DMAs done
        } else {
            asm volatile("s_wait_asynccnt 0" ::: "memory");
        }

        const char* base = s_stage + wave * 2048 + buf * 1024 + lane * 16;
        int4 s4 = *(const int4*)(base);          // ds_load_b128
        int4 d4 = *(const int4*)(base + 512);    // ds_load_b128

        // 4 edges * 3 floats of shift = 48B, 16B-aligned, coalesced
        const float* shp = shifts + (size_t)(t * 128 + lane * 4) * 3;
        float4 f0 = ((const float4*)shp)[0];
        float4 f1 = ((const float4*)shp)[1];
        float4 f2 = ((const float4*)shp)[2];

        zbl_edge_accum(coords, species, node_e, s4.x, d4.x, f0.x, f0.y, f0.z);
        zbl_edge_accum(coords, species, node_e, s4.y, d4.y, f0.w, f1.x, f1.y);
        zbl_edge_accum(coords, species, node_e, s4.z, d4.z, f1.z, f1.w, f2.x);
        zbl_edge_accum(coords, species, node_e, s4.w, d4.w, f2.y, f2.z, f2.w);
        buf ^= 1;
    }

    // Tail edges (E not multiple of 128): direct loads, no staging.
    for (int e = full_tiles * 128 + blockIdx.x * (int)blockDim.x + (int)threadIdx.x;
         e < E; e += gridDim.x * blockDim.x) {
        int s = ei[e];
        int d = ei[(size_t)E + e];
        const float* sp = shifts + 3 * (size_t)e;
        zbl_edge_accum(coords, species, node_e, s, d, sp[0], sp[1], sp[2]);
    }
}

// node_energy -> graph_energy segment sum. batch is sorted, G is small (256):
// accumulate in LDS bins with ds_add_f32, flush one global atomic per bin/block.
__global__ __launch_bounds__(256) void zbl_graph_kernel(
    const float* __restrict__ node_e, const int* __restrict__ batch,
    float* __restrict__ graph_e, int N, int G)
{
    __shared__ float bins[1024];
    const bool use_lds = (G <= 1024);
    if (use_lds) {
        for (int i = threadIdx.x; i < G; i += blockDim.x) bins[i] = 0.0f;
        __syncthreads();
    }
    for (int i = blockIdx.x * blockDim.x + threadIdx.x; i < N;
         i += gridDim.x * blockDim.x) {
        float v = node_e[i];
        int b = batch[i];
        if (use_lds) atomicAdd(&bins[b], v);            // ds_add_f32
        else         unsafeAtomicAdd(&graph_e[b], v);   // fallback
    }
    if (use_lds) {
        __syncthreads();
        for (int i = threadIdx.x; i < G; i += blockDim.x)
            unsafeAtomicAdd(&graph_e[i], bins[i]);
    }
}

extern "C" void kernel_launch(void* const* d_in, const int* in_sizes, int n_in,
                              void* d_out, int out_size, void* d_ws, size_t ws_size,
                              hipStream_t stream) {
    const float* coords  = (const float*)d_in[0];  // [N,3] f32
    const float* species = (const float*)d_in[1];  // [N]   f32
    const int*   ei      = (const int*)  d_in[2];  // [2,E] int
    const float* shifts  = (const float*)d_in[3];  // [E,3] f32
    const int*   batch   = (const int*)  d_in[4];  // [N]   int
    (void)d_ws; (void)ws_size; (void)n_in;

    const int N = in_sizes[1];
    const int E = in_sizes[2] / 2;
    const int G = out_size - N;

    float* node_e  = (float*)d_out;
    float* graph_e = node_e + N;

    // 1) zero outputs (harness poisons d_out)
    int tot = out_size;
    zbl_zero_kernel<<<(tot + 255) / 256, 256, 0, stream>>>((float*)d_out, tot);

    // 2) edge energies + node scatter (8192 waves in flight to hide L2 gather latency)
    zbl_edge_kernel<<<1024, 256, 0, stream>>>(coords, species, ei, shifts, node_e, E);

    // 3) node -> graph segment sum
    zbl_graph_kernel<<<256, 256, 0, stream>>>(node_e, batch, graph_e, N, G);
}